// GNNModule_32126355374294
// MI455X (gfx1250) — compile-verified
//
#include <hip/hip_runtime.h>
#include <hip/hip_bf16.h>

typedef float v2f __attribute__((ext_vector_type(2)));
typedef float v8f __attribute__((ext_vector_type(8)));

// ---------------- degree / normalization ----------------

__global__ __launch_bounds__(256) void init_deg(float* deg, int n) {
    int i = blockIdx.x * 256 + threadIdx.x;
    if (i < n) deg[i] = 1.0f;                 // self-loop contributes 1 to every node
}

__global__ __launch_bounds__(256) void deg_accum(const long long* __restrict__ dst,
                                                 float* deg, int nE) {
    int e = blockIdx.x * 256 + threadIdx.x;
    if (e < nE) atomicAdd(&deg[(int)dst[e]], 1.0f);
}

__global__ __launch_bounds__(256) void deg_to_dinv(float* deg, int n) {
    int i = blockIdx.x * 256 + threadIdx.x;
    if (i < n) {
        float d = deg[i];
        deg[i] = (d > 0.0f) ? rsqrtf(d) : 0.0f;
    }
}

// ---------------- bias broadcast (accumulator init) ----------------

__global__ __launch_bounds__(256) void init_bias(float* __restrict__ out,
                                                 const float* __restrict__ b,
                                                 int fMask, int total) {
    int i = blockIdx.x * 256 + threadIdx.x;
    if (i < total) out[i] = b[i & fMask];
}

// ---------------- dense GEMM via V_WMMA_F32_16X16X4_F32 ----------------
// Y[N,Kout] = op(X)[N,K] @ W[K,Kout], op = identity or ReLU (fused on load).
// grid = (N/16, Kout/128), block = 256 (8 waves); one 16x16 tile per wave.
template <int RELU>
__global__ __launch_bounds__(256) void gemm_wmma_f32(const float* __restrict__ X,
                                                     const float* __restrict__ W,
                                                     float* __restrict__ Y,
                                                     int K, int Kout) {
    const int lane    = threadIdx.x & 31;
    const int wave    = threadIdx.x >> 5;
    const int rowTile = blockIdx.x;
    const int colTile = blockIdx.y * 8 + wave;
    const int m       = lane & 15;            // row (A) / col (B) within tile
    const int kh      = (lane >> 4) << 1;     // 0 or 2: K sub-slot per half-wave

    const float* xp = X + (size_t)(rowTile * 16 + m) * K + kh;
    const float* wp = W + (size_t)kh * Kout + (colTile * 16 + m);

    v8f acc = {};
#pragma unroll 4
    for (int k = 0; k < K; k += 4) {
        v2f a, b;
        a.x = xp[0];
        a.y = xp[1];
        if (RELU) { a.x = fmaxf(a.x, 0.0f); a.y = fmaxf(a.y, 0.0f); }
        b.x = wp[0];
        b.y = wp[Kout];
        acc = __builtin_amdgcn_wmma_f32_16x16x4_f32(
            /*neg_a=*/false, a, /*neg_b=*/false, b,
            /*c_mod=*/(short)0, acc, /*reuse_a=*/false, /*reuse_b=*/false);
        xp += 4;
        wp += (size_t)Kout * 4;
    }

    // C/D layout: VGPR i -> M = i + (lane>>4)*8, N = lane&15
    const int rbase = rowTile * 16 + ((lane >> 4) << 3);
    float* yp = Y + (size_t)rbase * Kout + (colTile * 16 + m);
#pragma unroll
    for (int i = 0; i < 8; ++i)
        yp[(size_t)i * Kout] = acc[i];
}

// ---------------- edge scatter: Out[dst] += H[src] * dinv[src]*dinv[dst] ----------------
// One thread handles 4 consecutive features of one edge (float4 gather, 4 f32 atomics).
// Virtual edges [nE, nE+nN) are the self-loops.
__global__ __launch_bounds__(256) void scatter_edges(const float* __restrict__ Hm,
                                                     float* __restrict__ Out,
                                                     const long long* __restrict__ src,
                                                     const long long* __restrict__ dst,
                                                     const float* __restrict__ dinv,
                                                     int nE, int nN, int F, int fgShift) {
    const int fg  = 1 << fgShift;             // float4 groups per row (F/4)
    const int epb = 256 >> fgShift;           // edges per block
    const int e   = blockIdx.x * epb + (threadIdx.x >> fgShift);
    if (e >= nE + nN) return;
    const int f = (threadIdx.x & (fg - 1)) << 2;

    int s, d;
    if (e < nE) { s = (int)src[e]; d = (int)dst[e]; }
    else        { s = e - nE; d = s; }

    const float nrm = dinv[s] * dinv[d];
    const float4 h  = *(const float4*)(Hm + (size_t)s * F + f);
    float* op = Out + (size_t)d * F + f;
    atomicAdd(op + 0, h.x * nrm);
    atomicAdd(op + 1, h.y * nrm);
    atomicAdd(op + 2, h.z * nrm);
    atomicAdd(op + 3, h.w * nrm);
}

// ---------------- launch ----------------

extern "C" void kernel_launch(void* const* d_in, const int* in_sizes, int n_in,
                              void* d_out, int out_size, void* d_ws, size_t ws_size,
                              hipStream_t stream) {
    const float*     x  = (const float*)d_in[0];
    const long long* ei = (const long long*)d_in[1];   // int64 [2, E]
    const float*     W1 = (const float*)d_in[2];
    const float*     b1 = (const float*)d_in[3];
    const float*     W2 = (const float*)d_in[4];
    const float*     b2 = (const float*)d_in[5];

    const int H   = in_sizes[3];               // 256 (hidden)
    const int Fin = in_sizes[2] / H;           // 128 (input dim)
    const int N   = in_sizes[0] / Fin;         // 50000 nodes
    const int nE  = in_sizes[1] / 2;           // 800000 edges (w/o self-loops)
    const long long* src = ei;
    const long long* dst = ei + nE;

    // workspace: [dinv (N f32, padded) | bufA (N*H f32) | bufB (N*Fin f32)]
    char*  ws   = (char*)d_ws;
    float* dinv = (float*)ws;
    size_t off  = (((size_t)N * 4) + 255) & ~(size_t)255;
    float* bufA = (float*)(ws + off);                       // h1 / h2 / h3
    float* bufB = (float*)(ws + off + (size_t)N * H * 4);   // a2 (N x Fin)
    float* out  = (float*)d_out;                            // a1 (layer1) then final

    const int nTot = nE + N;                   // edges incl. self-loops
    auto scatterBlocks = [&](int F) {
        int fg  = F >> 2;
        int epb = 256 / fg;
        return (nTot + epb - 1) / epb;
    };
    const int shH   = 31 - __builtin_clz(H >> 2);    // log2(F/4) for F=H
    const int shFin = 31 - __builtin_clz(Fin >> 2);  // log2(F/4) for F=Fin

    // --- normalization ---
    init_deg   <<<(N  + 255) / 256, 256, 0, stream>>>(dinv, N);
    deg_accum  <<<(nE + 255) / 256, 256, 0, stream>>>(dst, dinv, nE);
    deg_to_dinv<<<(N  + 255) / 256, 256, 0, stream>>>(dinv, N);

    // --- layer 1: h1 = x @ W1 ; out = scatter(h1) + b1 (ReLU deferred into GEMM2) ---
    gemm_wmma_f32<0><<<dim3(N / 16, H / 128), 256, 0, stream>>>(x, W1, bufA, Fin, H);
    init_bias<<<((N * H) + 255) / 256, 256, 0, stream>>>(out, b1, H - 1, N * H);
    scatter_edges<<<scatterBlocks(H), 256, 0, stream>>>(bufA, out, src, dst, dinv,
                                                        nE, N, H, shH);

    // --- layer 2: h2 = relu(out) @ W2 ; bufB = scatter(h2) + b2 ---
    gemm_wmma_f32<1><<<dim3(N / 16, Fin / 128), 256, 0, stream>>>(out, W2, bufA, H, Fin);
    init_bias<<<((N * Fin) + 255) / 256, 256, 0, stream>>>(bufB, b2, Fin - 1, N * Fin);
    scatter_edges<<<scatterBlocks(Fin), 256, 0, stream>>>(bufA, bufB, src, dst, dinv,
                                                          nE, N, Fin, shFin);

    // --- layer 3: h3 = bufB @ W1 ; out = scatter(h3) + b1 ---
    gemm_wmma_f32<0><<<dim3(N / 16, H / 128), 256, 0, stream>>>(bufB, W1, bufA, Fin, H);
    init_bias<<<((N * H) + 255) / 256, 256, 0, stream>>>(out, b1, H - 1, N * H);
    scatter_edges<<<scatterBlocks(H), 256, 0, stream>>>(bufA, out, src, dst, dinv,
                                                        nE, N, H, shH);
}